// ATSSPostProcessor_8873402433638
// MI455X (gfx1250) — compile-verified
//
#include <hip/hip_runtime.h>
#include <hip/hip_bf16.h>

// ---- problem constants (match reference) ----
#define NIMG   16
#define HH     160
#define WWID   160
#define CC     80
#define HW     25600          // H*W
#define EPI    2048000        // C*H*W per image
#define PREK   1000
#define POSTK  200
#define CAP    4096           // compaction capacity per image
#define NMSTHR 0.8f
#define IMGSZ  1280.0f
#define TLOGIT (-2.9444389791664403f)   // logit(0.05)
#define CLIPLOG 4.135166556742356f      // log(1000/16)

// ---- workspace layout (byte offsets) ----
#define HIST_OFF   0u          // u32[16][2048]          = 131072
#define CNT_OFF    131072u     // u32[16]                = 64
#define THR_OFF    131136u     // u32[16]                = 64
#define CS_OFF     131200u     // f32[16][4096]          = 262144
#define CI_OFF     393344u     // u32[16][4096]          = 262144
#define NB_OFF     655488u     // f32[16][1024][4]       = 262144  (class-offset boxes)
#define BOX_OFF    917632u     // f32[16][1024][4]       = 262144  (output boxes)
#define SS_OFF     1179776u    // f32[16][1024]          = 65536   (sqrt scores)
#define SUP_OFF    1245312u    // u32[16][1024][32]      = 2097152 (suppression bitmask)

typedef float v2f __attribute__((ext_vector_type(2)));
typedef float v8f __attribute__((ext_vector_type(8)));

__device__ __forceinline__ float sigm(float x) { return 1.0f / (1.0f + expf(-x)); }

// ---------------- K0: zero a word range ----------------
__global__ void k_zero(unsigned* __restrict__ p, long nwords) {
  long i = (long)blockIdx.x * blockDim.x + threadIdx.x;
  long stride = (long)gridDim.x * blockDim.x;
  for (; i < nwords; i += stride) p[i] = 0u;
}

// ---------------- K1: per-image 2048-bin float-radix histogram ----------------
// grid (1000, 16), block 256; each thread handles 8 contiguous elems (2x b128 loads)
__global__ void k_hist(const float* __restrict__ cls, const float* __restrict__ ctr,
                       unsigned* __restrict__ hist) {
  __shared__ unsigned h[2048];
  const int n = blockIdx.y;
  for (int i = threadIdx.x; i < 2048; i += 256) h[i] = 0u;
  __syncthreads();
  const int e0 = blockIdx.x * 2048 + threadIdx.x * 8;      // first elem within image
  const long ibase = (long)n * EPI + e0;
  __builtin_prefetch(&cls[ibase + 2048], 0, 1);            // global_prefetch_b8
  float4 va = *(const float4*)(cls + ibase);
  float4 vb = *(const float4*)(cls + ibase + 4);
  float x[8] = {va.x, va.y, va.z, va.w, vb.x, vb.y, vb.z, vb.w};
#pragma unroll
  for (int j = 0; j < 8; ++j) {
    if (x[j] > TLOGIT) {
      int e = e0 + j;
      int hw = e % HW;
      float s = sigm(x[j]) * sigm(ctr[n * HW + hw]);       // s in (0,1): bits order-preserving
      unsigned key = __float_as_uint(s) >> 19;             // < 2048 bins
      atomicAdd(&h[key], 1u);
    }
  }
  __syncthreads();
  for (int i = threadIdx.x; i < 2048; i += 256)
    if (h[i]) atomicAdd(&hist[n * 2048 + i], h[i]);
}

// ---------------- K2: find top-1000 cutoff bin per image ----------------
__global__ void k_thresh(const unsigned* __restrict__ hist, unsigned* __restrict__ thr) {
  int n = threadIdx.x;
  if (n >= NIMG) return;
  unsigned cum = 0, T = 0;
  for (int k = 2047; k >= 0; --k) {
    cum += hist[n * 2048 + k];
    if (cum >= PREK) { T = (unsigned)k; break; }
  }
  thr[n] = T;                                              // key>=T passes (total<1000 -> T=0: take all)
}

// ---------------- K3: compact survivors (score, flat index) ----------------
__global__ void k_compact(const float* __restrict__ cls, const float* __restrict__ ctr,
                          const unsigned* __restrict__ thr, unsigned* __restrict__ cnt,
                          float* __restrict__ cs, unsigned* __restrict__ ci) {
  const int n = blockIdx.y;
  const unsigned T = thr[n];
  const int e0 = blockIdx.x * 2048 + threadIdx.x * 8;
  const long ibase = (long)n * EPI + e0;
  float4 va = *(const float4*)(cls + ibase);
  float4 vb = *(const float4*)(cls + ibase + 4);
  float x[8] = {va.x, va.y, va.z, va.w, vb.x, vb.y, vb.z, vb.w};
#pragma unroll
  for (int j = 0; j < 8; ++j) {
    if (x[j] > TLOGIT) {
      int e = e0 + j;
      int hw = e % HW;
      int c  = e / HW;
      float s = sigm(x[j]) * sigm(ctr[n * HW + hw]);
      if ((__float_as_uint(s) >> 19) >= T) {
        unsigned p = atomicAdd(&cnt[n], 1u);
        if (p < CAP) {
          cs[n * CAP + p] = s;
          ci[n * CAP + p] = (unsigned)(hw * CC + c);       // reference flat = hw*C + c
        }
      }
    }
  }
}

// ---------------- K4: per-image bitonic sort (desc, tie by idx) + decode ----------------
// one block (256 threads) per image; candidates staged into LDS via async loads
__global__ void k_sortdecode(const float* __restrict__ cs_g, const unsigned* __restrict__ ci_g,
                             const unsigned* __restrict__ cnt,
                             const float* __restrict__ reg, const float* __restrict__ anchors,
                             float* __restrict__ nb, float* __restrict__ boxes,
                             float* __restrict__ ss) {
  __shared__ float    s_s[CAP];
  __shared__ unsigned s_i[CAP];
  const int n = blockIdx.x;
  const int t = threadIdx.x;
  unsigned count = cnt[n]; if (count > CAP) count = CAP;

  for (int k = t; k < CAP; k += 256) {
    if (k < (int)count) {
      unsigned lds_s = (unsigned)(unsigned long long)&s_s[k];
      unsigned long long ga_s = (unsigned long long)&cs_g[n * CAP + k];
      asm volatile("global_load_async_to_lds_b32 %0, %1, off" :: "v"(lds_s), "v"(ga_s) : "memory");
      unsigned lds_i = (unsigned)(unsigned long long)&s_i[k];
      unsigned long long ga_i = (unsigned long long)&ci_g[n * CAP + k];
      asm volatile("global_load_async_to_lds_b32 %0, %1, off" :: "v"(lds_i), "v"(ga_i) : "memory");
    } else {
      s_s[k] = -1.0f;
      s_i[k] = 0xFFFFFFFFu;
    }
  }
  asm volatile("s_wait_asynccnt 0x0" ::: "memory");
  __syncthreads();

  // bitonic sort: score desc, flat-index asc on ties (matches jax top_k stability;
  // also makes atomic-compaction order irrelevant -> deterministic output)
  for (unsigned kk = 2; kk <= CAP; kk <<= 1) {
    for (unsigned j = kk >> 1; j > 0; j >>= 1) {
      for (unsigned i = t; i < CAP; i += 256) {
        unsigned l = i ^ j;
        if (l > i) {
          float sa = s_s[i], sb = s_s[l];
          unsigned ia = s_i[i], ib = s_i[l];
          bool aFirst = (sa > sb) || (sa == sb && ia < ib);
          bool desc = ((i & kk) == 0);
          if (desc != aFirst) { s_s[i] = sb; s_s[l] = sa; s_i[i] = ib; s_i[l] = ia; }
        }
      }
      __syncthreads();
    }
  }

  // decode top 1000 (pad to 1024 for 16x16 NMS tiling)
  for (int k = t; k < 1024; k += 256) {
    float sc = s_s[k];
    unsigned flat = s_i[k];
    float b0 = 0.f, b1 = 0.f, b2 = 0.f, b3 = 0.f, off = 0.f, outs = 0.f;
    if (k < PREK && sc > 0.0f) {
      int loc = (int)(flat / CC);
      int c   = (int)(flat % CC);
      float a0 = anchors[loc * 4 + 0], a1 = anchors[loc * 4 + 1];
      float a2 = anchors[loc * 4 + 2], a3 = anchors[loc * 4 + 3];
      float w = a2 - a0 + 1.0f, h = a3 - a1 + 1.0f;
      float cx = a0 + 0.5f * w, cy = a1 + 0.5f * h;
      const float* r = reg + (long)n * 4 * HW;
      float dx = r[0 * HW + loc] * 0.1f;
      float dy = r[1 * HW + loc] * 0.1f;
      float dw = fminf(r[2 * HW + loc] * 0.2f, CLIPLOG);
      float dh = fminf(r[3 * HW + loc] * 0.2f, CLIPLOG);
      float pcx = dx * w + cx, pcy = dy * h + cy;
      float pw = expf(dw) * w, ph = expf(dh) * h;
      b0 = fminf(fmaxf(pcx - 0.5f * pw, 0.f), IMGSZ - 1.f);
      b1 = fminf(fmaxf(pcy - 0.5f * ph, 0.f), IMGSZ - 1.f);
      b2 = fminf(fmaxf(pcx + 0.5f * pw - 1.f, 0.f), IMGSZ - 1.f);
      b3 = fminf(fmaxf(pcy + 0.5f * ph - 1.f, 0.f), IMGSZ - 1.f);
      off = (float)(c + 1) * (IMGSZ + 1.0f);               // ml_nms class offset
      outs = sqrtf(sc);
    }
    long o = (long)(n * 1024 + k);
    boxes[o * 4 + 0] = b0; boxes[o * 4 + 1] = b1; boxes[o * 4 + 2] = b2; boxes[o * 4 + 3] = b3;
    nb[o * 4 + 0] = b0 + off; nb[o * 4 + 1] = b1 + off; nb[o * 4 + 2] = b2 + off; nb[o * 4 + 3] = b3 + off;
    ss[n * 1024 + k] = outs;
  }
}

// ---------------- K6: suppression bitmask via 16x16 tiles; WMMA rank-2 area-sum ----------------
// grid (2080 upper-tri tiles, 16 images), block = 32 (one wave per tile)
__global__ void k_supmat(const float* __restrict__ nb, unsigned* __restrict__ sup) {
  const int n = blockIdx.y;
  int t = blockIdx.x, ti = 0, rem = 64;
  while (t >= rem) { t -= rem; ++ti; --rem; }              // upper-triangular unrank
  const int tj = ti + t;
  const int i0 = ti * 16, j0 = tj * 16;
  const int lane = threadIdx.x;
  const int lp = lane & 15;
  const bool lo = lane < 16;

  const float4* nb4 = (const float4*)(nb + (long)n * 1024 * 4);
  float4 cb = nb4[j0 + lp];                                // this lane's column box
  float areac = fmaxf(cb.z - cb.x, 0.f) * fmaxf(cb.w - cb.y, 0.f);
  float4 ab = nb4[i0 + lp];                                // A-operand row box (M = lane%16)
  float areaa = fmaxf(ab.z - ab.x, 0.f) * fmaxf(ab.w - ab.y, 0.f);

  // D[m][n] = area_row[m]*1 + 1*area_col[n]  (A 16x4: K0=area,K1=1; B 4x16: K0=1,K1=area)
  v2f a; a.x = lo ? areaa : 0.0f; a.y = lo ? 1.0f : 0.0f;
  v2f b; b.x = lo ? 1.0f : 0.0f;  b.y = lo ? areac : 0.0f;
  v8f c = {};
  c = __builtin_amdgcn_wmma_f32_16x16x4_f32(false, a, false, b, (short)0, c, false, false);

  const int rbase = i0 + (lo ? 0 : 8);                     // C-layout row group for this lane
  const unsigned myj = (unsigned)(j0 + lp);
#pragma unroll
  for (int k = 0; k < 8; ++k) {
    float4 rb = nb4[rbase + k];
    float iw = fminf(rb.z, cb.z) - fmaxf(rb.x, cb.x);
    float ih = fminf(rb.w, cb.w) - fmaxf(rb.y, cb.y);
    float inter = fmaxf(iw, 0.f) * fmaxf(ih, 0.f);
    float uni = fmaxf(c[k] - inter, 1e-9f);                // c[k] = area_i + area_j from WMMA
    // inter/uni > thr  <=>  inter > thr*uni  (uni > 0): division-free
    bool sup_k = (inter > NMSTHR * uni) && (myj > (unsigned)(rbase + k));
    unsigned long long bal = __ballot(sup_k);              // low16: rows 0-7, high16: rows 8-15
    if (lane == 0) {
      unsigned lob = (unsigned)(bal & 0xFFFFu);
      unsigned hib = (unsigned)((bal >> 16) & 0xFFFFu);
      int shift = j0 & 31;                                 // 0 or 16
      int wI = j0 >> 5;
      if (lob) atomicOr(&sup[((long)n * 1024 + (i0 + k)) * 32 + wI], lob << shift);
      if (hib) atomicOr(&sup[((long)n * 1024 + (i0 + 8 + k)) * 32 + wI], hib << shift);
    }
  }
}

// ---------------- K7: sequential bitmask NMS scan + top-200 output ----------------
// grid 16, block 32: lane owns one 32-bit keep word (1024 boxes)
__global__ void k_nms_out(const float* __restrict__ ss, const unsigned* __restrict__ sup,
                          const float* __restrict__ boxes, float* __restrict__ out) {
  const int n = blockIdx.x;
  const int lane = threadIdx.x;
  unsigned keepw = 0u;
  for (int b = 0; b < 32; ++b) {
    int idx = lane * 32 + b;
    if (idx < PREK && ss[n * 1024 + idx] > 0.f) keepw |= (1u << b);
  }
  for (int i = 0; i < PREK; ++i) {
    unsigned w = __shfl(keepw, i >> 5, 32);
    if ((w >> (i & 31)) & 1u)
      keepw &= ~sup[((long)n * 1024 + i) * 32 + lane];
  }
  // exclusive prefix of kept counts across lanes -> output rank (score-sorted order preserved)
  unsigned pc = __popc(keepw);
  unsigned incl = pc;
  for (int d = 1; d < 32; d <<= 1) {
    unsigned v = __shfl_up(incl, d, 32);
    if (lane >= d) incl += v;
  }
  unsigned excl = incl - pc;
  unsigned w = keepw;
  while (w) {
    int b = __ffs(w) - 1; w &= (w - 1u);
    unsigned r = excl + __popc(keepw & ((1u << b) - 1u));
    if (r < POSTK) {
      int idx = lane * 32 + b;
      float* o = out + ((long)n * POSTK + r) * 5;
      o[0] = boxes[((long)n * 1024 + idx) * 4 + 0];
      o[1] = boxes[((long)n * 1024 + idx) * 4 + 1];
      o[2] = boxes[((long)n * 1024 + idx) * 4 + 2];
      o[3] = boxes[((long)n * 1024 + idx) * 4 + 3];
      o[4] = ss[n * 1024 + idx];
    }
  }
}

extern "C" void kernel_launch(void* const* d_in, const int* in_sizes, int n_in,
                              void* d_out, int out_size, void* d_ws, size_t ws_size,
                              hipStream_t stream) {
  const float* cls = (const float*)d_in[0];   // [16,80,160,160]
  const float* reg = (const float*)d_in[1];   // [16,4,160,160]
  const float* ctr = (const float*)d_in[2];   // [16,1,160,160]
  const float* anc = (const float*)d_in[3];   // [25600,4]
  float* out = (float*)d_out;                 // [16,200,5]

  char* ws = (char*)d_ws;
  unsigned* hist  = (unsigned*)(ws + HIST_OFF);
  unsigned* cnt   = (unsigned*)(ws + CNT_OFF);
  unsigned* thr   = (unsigned*)(ws + THR_OFF);
  float*    cs    = (float*)   (ws + CS_OFF);
  unsigned* ci    = (unsigned*)(ws + CI_OFF);
  float*    nb    = (float*)   (ws + NB_OFF);
  float*    boxes = (float*)   (ws + BOX_OFF);
  float*    ss    = (float*)   (ws + SS_OFF);
  unsigned* sup   = (unsigned*)(ws + SUP_OFF);

  // deterministic re-init every call
  k_zero<<<128, 256, 0, stream>>>(hist, (THR_OFF + 64 - HIST_OFF) / 4);   // hist+cnt+thr
  k_zero<<<512, 256, 0, stream>>>(sup, (long)NIMG * 1024 * 32);
  k_zero<<<64, 256, 0, stream>>>((unsigned*)out, (long)NIMG * POSTK * 5);

  k_hist<<<dim3(1000, NIMG), 256, 0, stream>>>(cls, ctr, hist);
  k_thresh<<<1, 32, 0, stream>>>(hist, thr);
  k_compact<<<dim3(1000, NIMG), 256, 0, stream>>>(cls, ctr, thr, cnt, cs, ci);
  k_sortdecode<<<NIMG, 256, 0, stream>>>(cs, ci, cnt, reg, anc, nb, boxes, ss);
  k_supmat<<<dim3(2080, NIMG), 32, 0, stream>>>(nb, sup);
  k_nms_out<<<NIMG, 32, 0, stream>>>(ss, sup, boxes, out);
}